// BipartiteGCNLayer_712964571490
// MI455X (gfx1250) — compile-verified
//
#include <hip/hip_runtime.h>

typedef __attribute__((ext_vector_type(2))) float v2f;
typedef __attribute__((ext_vector_type(8))) float v8f;

#define DIN   128
#define DOUT  128
#define LDW_STRIDE 136   // W rows padded: 2nd half-wave hits disjoint bank group
#define LDA_STRIDE 132   // A rows padded: (4*m + k) % 64 distinct across lanes 0-15

// ---------------------------------------------------------------------------
// Kernel 1: WH = H_src @ W + b via V_WMMA_F32_16X16X4_F32
// Block: 256 threads (8 waves). Block computes rows [16*bid, 16*bid+16) x all
// 128 output columns; wave w owns the 16x16 tile at columns [16w, 16w+16).
// ---------------------------------------------------------------------------
__global__ void gcn_gemm_wmma(const float* __restrict__ H,
                              const float* __restrict__ W,
                              const float* __restrict__ bias,
                              float* __restrict__ WH,
                              int n_src)
{
    extern __shared__ float smem[];
    float* ldsW = smem;                      // 128 x LDW_STRIDE
    float* ldsA = smem + DIN * LDW_STRIDE;   // 16  x LDA_STRIDE

    const int tid  = threadIdx.x;
    const int row0 = blockIdx.x * 16;

    // Cooperative load of W (128x128) into LDS, padded stride.
    for (int i = tid; i < (DIN * DOUT) / 4; i += 256) {
        const int r  = i >> 5;          // 32 float4 per row
        const int c4 = (i & 31) << 2;
        const float4 v = ((const float4*)(W + (size_t)r * DOUT))[i & 31];
        float* dst = ldsW + r * LDW_STRIDE + c4;
        dst[0] = v.x; dst[1] = v.y; dst[2] = v.z; dst[3] = v.w;
    }
    // Cooperative load of the 16x128 A tile (shared by all 8 waves).
    for (int i = tid; i < (16 * DIN) / 4; i += 256) {
        const int r  = i >> 5;
        const int c4 = (i & 31) << 2;
        int row = row0 + r;
        if (row >= n_src) row = n_src - 1;   // clamp (n_src % 16 == 0 normally)
        const float4 v = ((const float4*)(H + (size_t)row * DIN))[i & 31];
        float* dst = ldsA + r * LDA_STRIDE + c4;
        dst[0] = v.x; dst[1] = v.y; dst[2] = v.z; dst[3] = v.w;
    }
    __syncthreads();

    const int wave = tid >> 5;
    const int lane = tid & 31;
    const int m    = lane & 15;            // A: row M, B/C/D: col N within tile
    const int khi  = (lane >> 4) << 1;     // lanes 0-15 -> K+{0,1}; 16-31 -> K+{2,3}
    const int n0   = wave * 16;

    v8f c = {};
    #pragma unroll
    for (int k0 = 0; k0 < DIN; k0 += 4) {
        v2f a, b;
        a.x = ldsA[m * LDA_STRIDE + k0 + khi];
        a.y = ldsA[m * LDA_STRIDE + k0 + khi + 1];
        b.x = ldsW[(k0 + khi)     * LDW_STRIDE + n0 + m];
        b.y = ldsW[(k0 + khi + 1) * LDW_STRIDE + n0 + m];
        // 8 args: (neg_a, A, neg_b, B, c_mod, C, reuse_a, reuse_b)
        c = __builtin_amdgcn_wmma_f32_16x16x4_f32(
                false, a, false, b, (short)0, c, false, false);
    }

    // C/D layout: VGPR j -> M = j (lanes 0-15) or M = j+8 (lanes 16-31); N = lane&15.
    const float bv = bias[n0 + m];
    const int mb   = (lane < 16) ? 0 : 8;
    #pragma unroll
    for (int j = 0; j < 8; ++j) {
        const int row = row0 + mb + j;
        if (row < n_src)
            WH[(size_t)row * DOUT + n0 + m] = c[j] + bv;
    }
}

// ---------------------------------------------------------------------------
// Kernel 2: zero agg (d_out) and deg (d_ws) -- harness poisons buffers.
// ---------------------------------------------------------------------------
__global__ void gcn_zero(float* __restrict__ agg, float* __restrict__ deg,
                         int n_agg, int n_deg)
{
    const int i = blockIdx.x * blockDim.x + threadIdx.x;
    if (i < n_agg) agg[i] = 0.0f;
    if (i < n_deg) deg[i] = 0.0f;
}

// ---------------------------------------------------------------------------
// Kernel 3: edge scatter. 32 lanes per edge; each lane moves a float4 of the
// WH row (coalesced 512B gather, served from L2) and does 4 atomic f32 adds.
// ---------------------------------------------------------------------------
__global__ void gcn_scatter(const float* __restrict__ WH,
                            const int* __restrict__ esrc,
                            const int* __restrict__ edst,
                            float* __restrict__ agg,
                            float* __restrict__ deg,
                            int n_edges)
{
    const int gid  = blockIdx.x * blockDim.x + threadIdx.x;
    const int edge = gid >> 5;
    const int lane = threadIdx.x & 31;
    if (edge >= n_edges) return;

    const int s = esrc[edge];
    const int d = edst[edge];

    const float4 v = ((const float4*)(WH + (size_t)s * DOUT))[lane];
    float* dst = agg + (size_t)d * DOUT + lane * 4;
    atomicAdd(dst + 0, v.x);
    atomicAdd(dst + 1, v.y);
    atomicAdd(dst + 2, v.z);
    atomicAdd(dst + 3, v.w);
    if (lane == 0) atomicAdd(deg + d, 1.0f);
}

// ---------------------------------------------------------------------------
// Kernel 4: out = relu(agg / max(deg,1)) in place on d_out.
// ---------------------------------------------------------------------------
__global__ void gcn_norm(float* __restrict__ agg, const float* __restrict__ deg,
                         int n_elems)
{
    const int i = blockIdx.x * blockDim.x + threadIdx.x;
    if (i >= n_elems) return;
    const float dg = fmaxf(deg[i >> 7], 1.0f);
    const float v  = agg[i] / dg;
    agg[i] = v > 0.0f ? v : 0.0f;
}

// ---------------------------------------------------------------------------
extern "C" void kernel_launch(void* const* d_in, const int* in_sizes, int n_in,
                              void* d_out, int out_size, void* d_ws, size_t ws_size,
                              hipStream_t stream)
{
    const float* H    = (const float*)d_in[0];   // [n_src, 128]
    const float* W    = (const float*)d_in[1];   // [128, 128]
    const float* bias = (const float*)d_in[2];   // [128]
    const int*   esrc = (const int*)d_in[3];     // [E]
    const int*   edst = (const int*)d_in[4];     // [E]

    const int n_src = in_sizes[0] / DIN;
    const int E     = in_sizes[3];
    const int n_dst = out_size / DOUT;

    // Workspace layout: WH [n_src*128] floats, then deg [n_dst] floats.
    float* WH  = (float*)d_ws;
    float* deg = (float*)((char*)d_ws + (size_t)n_src * DIN * sizeof(float));
    float* agg = (float*)d_out;

    // 1) GEMM with fp32 WMMA
    const int    gemm_blocks = (n_src + 15) / 16;
    const size_t shm = (size_t)(DIN * LDW_STRIDE + 16 * LDA_STRIDE) * sizeof(float);
    gcn_gemm_wmma<<<gemm_blocks, 256, shm, stream>>>(H, W, bias, WH, n_src);

    // 2) zero agg + deg
    const int n_agg = n_dst * DOUT;
    gcn_zero<<<(n_agg + 255) / 256, 256, 0, stream>>>(agg, deg, n_agg, n_dst);

    // 3) edge scatter (32 lanes per edge)
    const long long sth = (long long)E * 32;
    gcn_scatter<<<(int)((sth + 255) / 256), 256, 0, stream>>>(WH, esrc, edst, agg, deg, E);

    // 4) degree-normalize + ReLU
    gcn_norm<<<(n_agg + 255) / 256, 256, 0, stream>>>(agg, deg, n_agg);
}